// AddShift_mp_module_60035052863993
// MI455X (gfx1250) — compile-verified
//
#include <hip/hip_runtime.h>

// Problem constants (fixed by the reference module)
#define HIN       58
#define WIN       58
#define HOUT      56
#define WOUT      56
#define NKC       5            // NK: input channels per output channel
#define GIN       4            // GROUP_IN
#define COUT      128
#define CIN       640          // NKC * COUT
#define TILE_H    28           // HOUT split into 2 tiles so LDS fits in 64 KB
#define ROWSLOTS  40           // rows h0-7 .. h0+32 (shift range [-8,+4] applied to h+1)
#define ROWSTRIDE 70           // 8 left pad + 58 data + 4 right pad (floats) -> 8B-aligned rows
#define PLANE     (ROWSLOTS * ROWSTRIDE)   // 2800 floats per channel
#define LDS_FLOATS (NKC * PLANE)           // 14000 floats = 56,000 bytes
#define NTHREADS  224          // 7 wave32; 224*7 == 28*56 exactly (no tails)
#define NPOS      7

__global__ __launch_bounds__(NTHREADS)
void addshift_kernel(const float* __restrict__ x,
                     const int*   __restrict__ pad_hv,      // (640,8): cols 0..3 h-shifts, 4..7 v-shifts
                     const int*   __restrict__ idx_identit, // (128,4): channel ids in [5co,5co+5)
                     float* __restrict__ out, int Bx)
{
    __shared__ __align__(16) float lds[LDS_FLOATS];

    const int ht  = blockIdx.x;          // 0..1
    const int co  = blockIdx.y;          // 0..127
    const int b   = blockIdx.z;
    const int tid = threadIdx.x;

    const int c0    = co * NKC;
    const int h0    = ht * TILE_H;
    const int rbase = h0 - 7;                           // global row mapped to slot 0
    const int rlo   = (rbase < 0) ? 0 : rbase;
    const int rhi   = (h0 + 33 > HIN) ? HIN : h0 + 33;
    const int nrows = rhi - rlo;

    // ---- 1) selective zero: only bytes that are read but never written ----
    // valid row slot : cols 0..7 (left pad, H-gather) and 66..69 (right pad, H-gather)
    // invalid slot   : whole row (V-gather reads it at data columns)
    // data cols 8..65 of valid slots are fully overwritten by the async copy.
    {
        const int lo_inv  = rlo - rbase;     // slots [0, lo_inv) are invalid (tile 0: 7, tile 1: 0)
        const int hi_inv0 = rhi - rbase;     // slots [hi_inv0, 40) are invalid (tile 0: 40, tile 1: 37)
        float2* l2 = (float2*)lds;
        const float2 z2 = make_float2(0.f, 0.f);
        for (int i = tid; i < NKC * ROWSLOTS; i += NTHREADS) {   // 200 items, <=1 per thread
            int k  = i / ROWSLOTS;
            int rs = i - k * ROWSLOTS;
            int base2 = (k * PLANE + rs * ROWSTRIDE) >> 1;       // row start in float2 units (35/row)
            if (rs < lo_inv || rs >= hi_inv0) {
#pragma unroll
                for (int f = 0; f < ROWSTRIDE / 2; ++f) l2[base2 + f] = z2;  // whole empty row
            } else {
                l2[base2 + 0] = z2; l2[base2 + 1] = z2;          // cols 0..3
                l2[base2 + 2] = z2; l2[base2 + 3] = z2;          // cols 4..7
                l2[base2 + 33] = z2; l2[base2 + 34] = z2;        // cols 66..69
            }
        }
    }

    // ---- uniform shift / identity tables (scalar loads; blockIdx-uniform) ----
    int sh[NKC][GIN], sv[NKC][GIN], kid[GIN];
    {
        const int* ph = pad_hv + c0 * 8;
#pragma unroll
        for (int k = 0; k < NKC; ++k)
#pragma unroll
            for (int g = 0; g < GIN; ++g) {
                sh[k][g] = ph[k * 8 + g];
                sv[k][g] = ph[k * 8 + 4 + g];
            }
        const int* pid = idx_identit + co * GIN;
#pragma unroll
        for (int g = 0; g < GIN; ++g)
            kid[g] = pid[g] - c0;       // 0..4
    }

    // ---- dedup shifts into count-weights (wave-uniform; shift values are 4-3*si) ----
    float cH[NKC][NKC], cV[NKC][NKC], cI[NKC];
#pragma unroll
    for (int k = 0; k < NKC; ++k)
#pragma unroll
        for (int si = 0; si < NKC; ++si) {
            const int s = 4 - 3 * si;
            float a = 0.f, v = 0.f;
#pragma unroll
            for (int g = 0; g < GIN; ++g) {
                a += (sh[k][g] == s) ? 1.f : 0.f;
                v += (sv[k][g] == s) ? 1.f : 0.f;
            }
            cH[k][si] = a; cV[k][si] = v;
        }
#pragma unroll
    for (int c = 0; c < NKC; ++c) {
        float a = 0.f;
#pragma unroll
        for (int g = 0; g < GIN; ++g) a += (kid[g] == c) ? 1.f : 0.f;
        cI[c] = a;
    }

    __syncthreads();   // pad zeroing complete before async engine writes LDS

    // ---- 2) async B64 copy: 5 channel planes -> padded LDS tiles ----
    // GVS mode: uniform 64-bit SGPR base + per-lane 32-bit byte offset.
    // Row stride in mem = 232 B, in LDS = 280 B, data starts at +32 B: all 8B-aligned.
    const float*   gbase    = x + ((size_t)b * CIN + c0) * (HIN * WIN);
    const unsigned lds_base = (unsigned)(size_t)lds;  // low 32 bits of flat ptr == LDS offset
    const int      per_ch   = nrows * (WIN / 2);      // 29 float2 per row

    for (int k = 0; k < NKC; ++k) {
        const int      gch = k * (HIN * WIN * 4);
        const unsigned lch = lds_base + (unsigned)(k * PLANE * 4);
        for (int i = tid; i < per_ch; i += NTHREADS) {
            int rr = i / 29;                // division by constant -> mul
            int w2 = i - rr * 29;
            int r  = rlo + rr;
            int      voff  = gch + r * (WIN * 4) + w2 * 8;
            unsigned laddr = lch + (unsigned)((r - rbase) * (ROWSTRIDE * 4) + 32 + w2 * 8);
            asm volatile("global_load_async_to_lds_b64 %0, %1, %2"
                         :
                         : "v"(laddr), "v"(voff), "s"(gbase)
                         : "memory");
        }
    }
    asm volatile("s_wait_asynccnt 0x0" ::: "memory");
    __syncthreads();   // all waves' async transfers complete

    // ---- 3) gather + reduce out of LDS ----
    // position idx = tid + it*224; row slot for input row h+1 is hl+8, col slot for w+1 is 9+w
    int rowb[NPOS];
#pragma unroll
    for (int it = 0; it < NPOS; ++it) {
        int idx = tid + it * NTHREADS;
        int hl  = idx / WOUT;
        int w   = idx - hl * WOUT;
        rowb[it] = hl * ROWSTRIDE + 9 + w;    // +8 row bias folded into constant offsets below
    }

    float accH[NPOS] = {}, accV[NPOS] = {}, accI[NPOS] = {};
#pragma unroll
    for (int k = 0; k < NKC; ++k) {
#pragma unroll
        for (int si = 0; si < NKC; ++si) {
            const int s = 4 - 3 * si;
            const float a = cH[k][si];
            if (a != 0.f) {            // uniform branch (scalar)
#pragma unroll
                for (int it = 0; it < NPOS; ++it)
                    accH[it] = fmaf(a, lds[k * PLANE + 8 * ROWSTRIDE + s + rowb[it]], accH[it]);
            }
            const float v = cV[k][si];
            if (v != 0.f) {            // uniform branch (scalar)
#pragma unroll
                for (int it = 0; it < NPOS; ++it)
                    accV[it] = fmaf(v, lds[k * PLANE + (8 + s) * ROWSTRIDE + rowb[it]], accV[it]);
            }
        }
    }
#pragma unroll
    for (int c = 0; c < NKC; ++c) {
        const float a = cI[c];
        if (a != 0.f) {                // uniform branch (scalar)
#pragma unroll
            for (int it = 0; it < NPOS; ++it)
                accI[it] = fmaf(a, lds[c * PLANE + 8 * ROWSTRIDE + rowb[it]], accI[it]);
        }
    }

    // ---- 4) stream results (write-once -> non-temporal) ----
    const size_t N     = (size_t)Bx * COUT * HOUT * WOUT;
    const size_t obase = (((size_t)b * COUT + co) * HOUT + h0) * WOUT;
    float* oH = out + obase;
    float* oV = out + N + obase;
    float* oI = out + 2 * N + obase;
#pragma unroll
    for (int it = 0; it < NPOS; ++it) {
        const int o = tid + it * NTHREADS;   // == hl*WOUT + w
        __builtin_nontemporal_store(accH[it], oH + o);
        __builtin_nontemporal_store(accV[it], oV + o);
        __builtin_nontemporal_store(accI[it], oI + o);
    }
}

extern "C" void kernel_launch(void* const* d_in, const int* in_sizes, int n_in,
                              void* d_out, int out_size, void* d_ws, size_t ws_size,
                              hipStream_t stream) {
    const float* x           = (const float*)d_in[0];
    const int*   pad_hv      = (const int*)d_in[1];
    const int*   idx_identit = (const int*)d_in[2];
    // d_in[3]/d_in[4] are hout/wout scalars; constants compiled in.
    float* out = (float*)d_out;

    const int Bx = in_sizes[0] / (CIN * HIN * WIN);   // 32 for reference shapes

    dim3 grid(2, COUT, Bx);
    addshift_kernel<<<grid, NTHREADS, 0, stream>>>(x, pad_hv, idx_identit, out, Bx);
}